// PLPContrastiveLoss_27839978012947
// MI455X (gfx1250) — compile-verified
//
#include <hip/hip_runtime.h>
#include <math.h>

#define BB   32
#define TT   8192
#define DD   128
#define AA   64
#define KN   32
#define RSTR 132   // padded LDS row stride (floats) to dodge bank conflicts

typedef __attribute__((ext_vector_type(4))) int   v4i;
typedef __attribute__((ext_vector_type(2))) float v2f;
typedef __attribute__((ext_vector_type(8))) float v8f;

#if __has_builtin(__builtin_amdgcn_global_load_async_to_lds_b128) && \
    __has_builtin(__builtin_amdgcn_s_wait_asynccnt)
#define HAVE_ASYNC_LDS 1
typedef __attribute__((address_space(1))) v4i v4i_g;   // global (AS1) int4
typedef __attribute__((address_space(3))) v4i v4i_l;   // LDS    (AS3) int4
#endif

// ---------------- threefry2x32, exact JAX semantics (key = (0,42)) -------------
__device__ __forceinline__ unsigned rotl32(unsigned x, int n) {
    return (x << n) | (x >> (32 - n));
}

__device__ __forceinline__ float thrf_uniform(unsigned idx) {
    // jax.random.uniform(key(42), (B,A,T)): bits = threefry2x32((0,42), iota(size))
    const unsigned half = (unsigned)(BB * AA * TT) / 2u;  // 8388608
    unsigned j  = (idx < half) ? idx : (idx - half);
    unsigned x0 = j;
    unsigned x1 = half + j;
    const unsigned ks0 = 0u, ks1 = 42u;
    const unsigned ks2 = ks0 ^ ks1 ^ 0x1BD11BDAu;
    x0 += ks0; x1 += ks1;
    const int r0[4] = {13, 15, 26, 6};
    const int r1[4] = {17, 29, 16, 24};
#define TF_RB(rt) { x0 += x1; x1 = rotl32(x1, rt[0]); x1 ^= x0; \
                    x0 += x1; x1 = rotl32(x1, rt[1]); x1 ^= x0; \
                    x0 += x1; x1 = rotl32(x1, rt[2]); x1 ^= x0; \
                    x0 += x1; x1 = rotl32(x1, rt[3]); x1 ^= x0; }
    TF_RB(r0); x0 += ks1; x1 += ks2 + 1u;
    TF_RB(r1); x0 += ks2; x1 += ks0 + 2u;
    TF_RB(r0); x0 += ks0; x1 += ks1 + 3u;
    TF_RB(r1); x0 += ks1; x1 += ks2 + 4u;
    TF_RB(r0); x0 += ks2; x1 += ks0 + 5u;
#undef TF_RB
    unsigned bits = (idx < half) ? x0 : x1;
    unsigned u = (bits >> 9) | 0x3f800000u;
    return __uint_as_float(u) - 1.0f;  // [0,1)
}

// ---------------- K1: stable compaction of active peaks -----------------------
__global__ __launch_bounds__(256) void k_compact(const float* __restrict__ pm,
                                                 const float* __restrict__ pv,
                                                 const float* __restrict__ vm,
                                                 int* __restrict__ order,
                                                 float* __restrict__ w,
                                                 int* __restrict__ count) {
    int b = blockIdx.x;
    __shared__ int waveTot[8];
    __shared__ int waveOff[8];
    __shared__ int base;
    if (threadIdx.x == 0) base = 0;
    __syncthreads();
    int lane = threadIdx.x & 31;
    int wid  = threadIdx.x >> 5;
    for (int t0 = 0; t0 < TT; t0 += 256) {
        int t = t0 + (int)threadIdx.x;
        bool act = (pm[b * TT + t] > 0.0f) && (vm[b * TT + t] > 0.0f);
        unsigned long long m = __ballot(act ? 1 : 0);
        unsigned mlo = (unsigned)m;
        int rank = __popc(mlo & ((1u << lane) - 1u));
        if (lane == 0) waveTot[wid] = __popc(mlo);
        __syncthreads();
        if (threadIdx.x == 0) {
            int s = base;
            for (int i = 0; i < 8; i++) { waveOff[i] = s; s += waveTot[i]; }
            base = s;
        }
        __syncthreads();
        if (act) {
            int slot = waveOff[wid] + rank;
            order[b * TT + slot] = t;
            float x = pv[b * TT + t];
            w[b * TT + slot] = (x < 1e-6f) ? 1e-6f : x;
        }
        __syncthreads();
    }
    if (threadIdx.x == 0) count[b] = base;
}

// ---------------- K2: top-64 anchors by weight + positive selection -----------
__global__ __launch_bounds__(256) void k_anchor(const float* __restrict__ w,
                                                const int* __restrict__ count,
                                                int* __restrict__ anchors,
                                                int* __restrict__ positive) {
    int b = blockIdx.x;
    int tid = threadIdx.x;
    int cnt = count[b];
    __shared__ float sw[TT];
    __shared__ unsigned char taken[TT];
    __shared__ float bestV[256];
    __shared__ int   bestI[256];
    for (int s = tid; s < TT; s += 256) {
        taken[s] = 0;
        sw[s] = (s < cnt) ? w[b * TT + s] : -1.0f;
    }
    __syncthreads();
    int na = (cnt < AA) ? cnt : AA;
    for (int r = 0; r < AA; r++) {
        int slot;
        if (r < na) {
            float bv = -1.0f; int bi = TT;
            for (int s = tid; s < cnt; s += 256) {
                if (!taken[s]) {
                    float v = sw[s];
                    if (v > bv) { bv = v; bi = s; }   // ascending s: strict > keeps lowest index on tie
                }
            }
            bestV[tid] = bv; bestI[tid] = bi;
            __syncthreads();
            for (int off = 128; off > 0; off >>= 1) {
                if (tid < off) {
                    float ov = bestV[tid + off]; int oi = bestI[tid + off];
                    if (ov > bestV[tid] || (ov == bestV[tid] && oi < bestI[tid])) {
                        bestV[tid] = ov; bestI[tid] = oi;
                    }
                }
                __syncthreads();
            }
            slot = bestI[0];
            if (tid == 0 && slot < TT) taken[slot] = 1;
        } else {
            slot = r;  // argsort places -inf slots in ascending slot order
        }
        if (tid == 0) anchors[b * AA + r] = slot;
        __syncthreads();
    }
    // positives: argmax of w over anchor + {1,-1,2,-2,4,-4}, first-max tie-break
    if (tid < AA) {
        int as_ = anchors[b * AA + tid];
        const int OFF[6] = {1, -1, 2, -2, 4, -4};
        float bw = -1.0f; int bo = 0; bool any = false;
        for (int jj = 0; jj < 6; jj++) {
            int c = as_ + OFF[jj];
            if (c >= 0 && c < cnt) {
                float v = sw[c];
                if (!any || v > bw) { any = true; bw = v; bo = jj; }
            }
        }
        positive[b * AA + tid] = as_ + OFF[bo];  // all-invalid -> argmax 0 -> anchor+1 (matches jnp.argmax)
    }
}

// ---------------- K3: per-(b,a) negatives = top-32 of threefry scores ---------
__global__ __launch_bounds__(256) void k_negs(const int* __restrict__ count,
                                              const int* __restrict__ anchors,
                                              const int* __restrict__ positive,
                                              int* __restrict__ negs,
                                              int* __restrict__ negcnt) {
    int pair = blockIdx.x;
    int b = pair >> 6, a = pair & (AA - 1);
    int tid = threadIdx.x;
    int cnt = count[b];
    int nmin = (cnt < AA) ? cnt : AA;
    bool anchor_valid = (a < nmin) && (cnt >= 2);
    if (!anchor_valid) {
        if (tid == 0) negcnt[pair] = 0;
        return;
    }
    __shared__ float sc[TT];
    __shared__ float bestV[256];
    __shared__ int   bestI[256];
    __shared__ int   nprim;
    int as_ = anchors[pair];
    int ps  = positive[pair];
    if (tid == 0) nprim = 0;
    __syncthreads();
    int local = 0;
    for (int s = tid; s < cnt; s += 256) {
        int d = (s > as_) ? (s - as_) : (as_ - s);
        if (s != as_ && s != ps && d > 4) local++;
    }
    atomicAdd(&nprim, local);
    __syncthreads();
    bool usePrim = (nprim > 0);
    for (int s = tid; s < cnt; s += 256) {
        int d = (s > as_) ? (s - as_) : (as_ - s);
        bool base_ = (s != as_) && (s != ps);
        bool msk = usePrim ? (base_ && d > 4) : base_;
        sc[s] = msk ? thrf_uniform((unsigned)pair * (unsigned)TT + (unsigned)s) : -1.0f;
    }
    __syncthreads();
    int found = 0;
    for (int r = 0; r < KN; r++) {
        float bv = -1.0f; int bi = TT;
        for (int s = tid; s < cnt; s += 256) {
            float v = sc[s];
            if (v > bv) { bv = v; bi = s; }
        }
        bestV[tid] = bv; bestI[tid] = bi;
        __syncthreads();
        for (int off = 128; off > 0; off >>= 1) {
            if (tid < off) {
                float ov = bestV[tid + off]; int oi = bestI[tid + off];
                if (ov > bestV[tid] || (ov == bestV[tid] && oi < bestI[tid])) {
                    bestV[tid] = ov; bestI[tid] = oi;
                }
            }
            __syncthreads();
        }
        float bv0 = bestV[0]; int bi0 = bestI[0];
        if (bv0 < 0.0f) break;                       // fewer than 32 negatives available
        if (tid == 0) { negs[pair * KN + r] = bi0; sc[bi0] = -1.0f; }
        found++;
        __syncthreads();
    }
    if (tid == 0) negcnt[pair] = found;
}

// ---------------- K4: FP32 WMMA logits + masked logsumexp loss ----------------
__global__ __launch_bounds__(32) void k_loss(const float* __restrict__ emb,
                                             const int* __restrict__ order,
                                             const float* __restrict__ w,
                                             const int* __restrict__ count,
                                             const int* __restrict__ anchors,
                                             const int* __restrict__ positive,
                                             const int* __restrict__ negs,
                                             const int* __restrict__ negcnt,
                                             float* __restrict__ lossArr,
                                             float* __restrict__ waArr) {
    int pair = blockIdx.x;
    int b = pair >> 6, a = pair & (AA - 1);
    int lane = threadIdx.x;
    int cnt = count[b];
    int nn  = negcnt[pair];
    int nmin = (cnt < AA) ? cnt : AA;
    bool pv = (a < nmin) && (cnt >= 2) && (nn > 0);
    if (!pv) {
        if (lane == 0) { lossArr[pair] = 0.0f; waArr[pair] = 0.0f; }
        return;
    }
    __shared__ __align__(16) float rows[48 * RSTR];
    __shared__ __align__(16) float avec[DD];
    __shared__ float dots[48];
    for (int i = lane; i < 48 * RSTR; i += 32) rows[i] = 0.0f;
    // DS stores (DScnt) and async-to-LDS loads (ASYNCcnt) are UNORDERED:
    // barrier so the zero-fill lands before the async gathers overwrite rows 0..nn.
    __syncthreads();

    int as_ = anchors[pair];
    int ps  = positive[pair];
    const float* embB = emb + (size_t)b * TT * DD;

    // anchor vector: 32 lanes x 16B = 512B row
    {
        int aorig = order[b * TT + as_];
        const float* src = embB + (size_t)aorig * DD + lane * 4;
#ifdef HAVE_ASYNC_LDS
        __builtin_amdgcn_global_load_async_to_lds_b128(
            (v4i_g*)src, (v4i_l*)&avec[lane * 4], 0, 0);
#else
        __builtin_prefetch(src, 0, 1);               // global_prefetch_b8
        ((float4*)avec)[lane] = *(const float4*)src;
#endif
    }
    // row 0 = positive, rows 1..nn = negatives (slot -> orig via order)
    for (int r = 0; r <= nn; r++) {
        int slot = (r == 0) ? ps : negs[pair * KN + (r - 1)];
        int orig = order[b * TT + slot];
        const float* src = embB + (size_t)orig * DD + lane * 4;
#ifdef HAVE_ASYNC_LDS
        __builtin_amdgcn_global_load_async_to_lds_b128(
            (v4i_g*)src, (v4i_l*)&rows[r * RSTR + lane * 4], 0, 0);
#else
        __builtin_prefetch(src, 0, 1);
        ((float4*)&rows[r * RSTR])[lane] = *(const float4*)src;
#endif
    }
#ifdef HAVE_ASYNC_LDS
    __builtin_amdgcn_s_wait_asynccnt(0);             // s_wait_asynccnt 0
#endif
    __syncthreads();

    // D = Rows(16x4) x Abcast(4x16), accumulated over K=128 in 32 WMMA steps.
    // A layout (ISA 16x4 f32): lanes 0-15 M=lane {K0,K1}; lanes 16-31 M=lane-16 {K2,K3}.
    // B layout (4x16): VGPR i, lane-half h -> K = i + 2h; broadcast anchor over all 16 cols.
    int m15 = lane & 15;
    int kh  = lane >> 4;
    for (int tile = 0; tile < 3; tile++) {
        v8f c = {0.f, 0.f, 0.f, 0.f, 0.f, 0.f, 0.f, 0.f};
        const float* rbase = &rows[(tile * 16 + m15) * RSTR];
        for (int kk = 0; kk < DD; kk += 4) {
            int klo = kk + 2 * kh;
            v2f Av; Av.x = rbase[klo]; Av.y = rbase[klo + 1];
            v2f Bv; Bv.x = avec[klo];  Bv.y = avec[klo + 1];
            c = __builtin_amdgcn_wmma_f32_16x16x4_f32(false, Av, false, Bv,
                                                      (short)0, c, false, false);
        }
        // C col 0: lane 0 holds M=0..7 in c[0..7]; lane 16 holds M=8..15.
        if (lane == 0)  { for (int r = 0; r < 8; r++) dots[tile * 16 + r]     = c[r]; }
        if (lane == 16) { for (int r = 0; r < 8; r++) dots[tile * 16 + 8 + r] = c[r]; }
    }
    __syncthreads();

    if (lane == 0) {
        const float inv = 10.0f;                 // 1 / TEMP
        float l0 = dots[0] * inv;
        float m = l0;
        for (int i = 1; i <= nn; i++) m = fmaxf(m, dots[i] * inv);
        float ssum = 0.0f;
        for (int i = 0; i <= nn; i++) ssum += expf(dots[i] * inv - m);
        float loss = logf(ssum) + m - l0;
        float wa = w[b * TT + as_];
        lossArr[pair] = loss * wa;
        waArr[pair]   = wa;
    }
}

// ---------------- K5: deterministic weighted-mean reduction -------------------
__global__ __launch_bounds__(256) void k_final(const float* __restrict__ lossArr,
                                               const float* __restrict__ waArr,
                                               float* __restrict__ out) {
    __shared__ float sn[256], sd[256];
    int tid = threadIdx.x;
    float n = 0.0f, d = 0.0f;
    for (int i = tid; i < BB * AA; i += 256) { n += lossArr[i]; d += waArr[i]; }
    sn[tid] = n; sd[tid] = d;
    __syncthreads();
    for (int off = 128; off > 0; off >>= 1) {
        if (tid < off) { sn[tid] += sn[tid + off]; sd[tid] += sd[tid + off]; }
        __syncthreads();
    }
    if (tid == 0) out[0] = (sd[0] > 0.0f) ? (sn[0] / sd[0]) : 0.0f;
}

// ---------------- launch ------------------------------------------------------
extern "C" void kernel_launch(void* const* d_in, const int* in_sizes, int n_in,
                              void* d_out, int out_size, void* d_ws, size_t ws_size,
                              hipStream_t stream) {
    const float* emb = (const float*)d_in[0];
    const float* pm  = (const float*)d_in[1];
    const float* pvv = (const float*)d_in[2];
    const float* vm  = (const float*)d_in[3];
    float* out = (float*)d_out;

    char* ws = (char*)d_ws;
    size_t off = 0;
    auto carve = [&](size_t bytes) -> void* {
        void* p = ws + off;
        off += (bytes + 255) & ~(size_t)255;
        return p;
    };
    int*   order    = (int*)  carve((size_t)BB * TT * 4);
    float* w        = (float*)carve((size_t)BB * TT * 4);
    int*   count    = (int*)  carve((size_t)BB * 4);
    int*   anchors  = (int*)  carve((size_t)BB * AA * 4);
    int*   positive = (int*)  carve((size_t)BB * AA * 4);
    int*   negs     = (int*)  carve((size_t)BB * AA * KN * 4);
    int*   negcnt   = (int*)  carve((size_t)BB * AA * 4);
    float* lossArr  = (float*)carve((size_t)BB * AA * 4);
    float* waArr    = (float*)carve((size_t)BB * AA * 4);
    (void)in_sizes; (void)n_in; (void)out_size; (void)ws_size;

    k_compact<<<BB, 256, 0, stream>>>(pm, pvv, vm, order, w, count);
    k_anchor <<<BB, 256, 0, stream>>>(w, count, anchors, positive);
    k_negs   <<<BB * AA, 256, 0, stream>>>(count, anchors, positive, negs, negcnt);
    k_loss   <<<BB * AA, 32, 0, stream>>>(emb, order, w, count, anchors, positive,
                                          negs, negcnt, lossArr, waArr);
    k_final  <<<1, 256, 0, stream>>>(lossArr, waArr, out);
}